// OMRAbstractModel_46686294508151
// MI455X (gfx1250) — compile-verified
//
#include <hip/hip_runtime.h>
#include <stdint.h>

// ---------------- problem constants (from reference) ----------------
#define N_BOX     8192
#define NWORDS    (N_BOX / 32)       // 256 mask words per row
#define THR_F     0.7f
#define IOU_THR_F 0.5f
#define EPS_F     1e-9f
// SCALER = max(3508/1280, 2480/1280) = 2.740625
#define SCALER_F  2.740625f

// =====================================================================
// K1: masked score per box:  valid ? max(c0,c1) : -inf
// =====================================================================
__global__ void nms_score_kernel(const float* __restrict__ conf,
                                 float* __restrict__ msOut) {
    int i = blockIdx.x * 256 + threadIdx.x;
    float c0 = conf[2 * i + 0];
    float c1 = conf[2 * i + 1];
    bool valid = (c0 > THR_F) || (c1 > THR_F);
    float s = fmaxf(c0, c1);
    msOut[i] = valid ? s : -__builtin_inff();
}

// =====================================================================
// K2: single-workgroup LDS bitonic argsort of 8192 (score, idx) pairs.
// 64 KB of LDS — possible because CDNA5 gives 320 KB per WGP.
// Comparator: descending score, ties -> ascending index == stable
// argsort(-masked_scores).
// =====================================================================
__global__ __launch_bounds__(1024)
void nms_sort_kernel(const float* __restrict__ score,
                     int* __restrict__ idxOut) {
    __shared__ float k[N_BOX];
    __shared__ int   v[N_BOX];
    const int tid = threadIdx.x;            // 0..1023
    for (int i = tid; i < N_BOX; i += 1024) { k[i] = score[i]; v[i] = i; }
    __syncthreads();

    for (int size = 2; size <= N_BOX; size <<= 1) {
        for (int stride = size >> 1; stride > 0; stride >>= 1) {
            for (int t = tid; t < N_BOX / 2; t += 1024) {
                int lo = 2 * t - (t & (stride - 1));
                int hi = lo + stride;
                bool dirAsc = ((lo & size) == 0);
                float ka = k[lo], kb = k[hi];
                int   va = v[lo], vb = v[hi];
                bool b_before_a = (kb > ka) || ((kb == ka) && (vb < va));
                bool a_before_b = (ka > kb) || ((ka == kb) && (va < vb));
                bool doSwap = dirAsc ? b_before_a : a_before_b;
                if (doSwap) {
                    k[lo] = kb; k[hi] = ka;
                    v[lo] = vb; v[hi] = va;
                }
            }
            __syncthreads();
        }
    }
    for (int i = tid; i < N_BOX; i += 1024) idxOut[i] = v[i];
}

// =====================================================================
// K3: gather boxes/conf into sorted order; build initial keep bitmask
// with wave32 __ballot.  Block 0 also zero-inits rowAny (mask kernel
// atomically ORs into it; stream order guarantees init-before-use).
// =====================================================================
__global__ void nms_gather_kernel(const float* __restrict__ conf,
                                  const float* __restrict__ boxes,
                                  const int* __restrict__ order,
                                  float* __restrict__ sBoxes,
                                  float* __restrict__ sConf,
                                  unsigned* __restrict__ keepInit,
                                  unsigned* __restrict__ rowAny) {
    int i = blockIdx.x * 256 + threadIdx.x;
    if (blockIdx.x == 0) rowAny[threadIdx.x] = 0u;
    int o = order[i];
    float4 bb = ((const float4*)boxes)[o];
    ((float4*)sBoxes)[i] = bb;
    float c0 = conf[2 * o + 0];
    float c1 = conf[2 * o + 1];
    sConf[2 * i + 0] = c0;
    sConf[2 * i + 1] = c1;
    bool valid = (c0 > THR_F) || (c1 > THR_F);
    unsigned long long bal = __ballot(valid);     // wave32: low 32 bits
    if ((threadIdx.x & 31) == 0) keepInit[i >> 5] = (unsigned)bal;
}

// =====================================================================
// K4: suppression bitmask.  Block = 256 rows x 32 cols.  Column tile
// staged into LDS via CDNA5 async global->LDS DMA (ASYNCcnt path).
// Also emits:
//   diag[i]   = row i's suppression word against its OWN col chunk
//   rowAny    = bit i set iff row i suppresses anything BEYOND its word
// =====================================================================
__global__ __launch_bounds__(256)
void nms_mask_kernel(const float* __restrict__ sBoxes,
                     unsigned* __restrict__ mask,
                     unsigned* __restrict__ diag,
                     unsigned* __restrict__ rowAny) {
    __shared__ float colBox[32 * 4];

    const int tid   = threadIdx.x;
    const int cbase = blockIdx.x * 32;    // column chunk base (0..8160)
    const int rbase = blockIdx.y * 256;   // row tile base

    // --- async DMA of the 32 column boxes into LDS (wave 0 only) ---
    if (tid < 32) {
        unsigned ldsAddr = (unsigned)(uintptr_t)(&colBox[tid * 4]); // low 32b of flat = LDS offset
        unsigned long long gAddr =
            (unsigned long long)(uintptr_t)(sBoxes + (size_t)(cbase + tid) * 4);
        asm volatile("global_load_async_to_lds_b128 %0, %1, off"
                     :: "v"(ldsAddr), "v"(gAddr)
                     : "memory");
    }
    asm volatile("s_wait_asynccnt 0" ::: "memory");
    __syncthreads();

    const int r = rbase + tid;
    float4 rb = ((const float4*)sBoxes)[r];
    float areaR = (rb.z - rb.x) * (rb.w - rb.y);

    const float4* cb4 = (const float4*)colBox;
    unsigned word = 0u;
#pragma unroll
    for (int c = 0; c < 32; ++c) {
        float4 cb = cb4[c];
        int j = cbase + c;
        float iw = fminf(rb.z, cb.z) - fmaxf(rb.x, cb.x);
        float ih = fminf(rb.w, cb.w) - fmaxf(rb.y, cb.y);
        iw = fmaxf(iw, 0.0f);
        ih = fmaxf(ih, 0.0f);
        float inter = iw * ih;
        float areaC = (cb.z - cb.x) * (cb.w - cb.y);
        float iou = inter / (areaR + areaC - inter + EPS_F);
        if ((j > r) && (iou > IOU_THR_F)) word |= (1u << c);
    }
    mask[(size_t)r * NWORDS + blockIdx.x] = word;
    if (cbase == (r & ~31)) diag[r] = word;              // own-word row
    if ((cbase > r) && (word != 0u))                     // beyond own word
        atomicOr(&rowAny[r >> 5], 1u << (r & 31));
}

// =====================================================================
// K5: greedy suppression scan — SINGLE wave32 wave, zero barriers.
// The 256 "removed" words live in registers (8 per lane, lane-
// interleaved: rem[k] holds word k*32+lane).  Per 32-box word:
//   1. broadcast removed word via __shfl (ds_bpermute, wave32 path)
//   2. resolve intra-word greedy chain purely on registers using the
//      precomputed diag words (shuffled from the owning lane)
//   3. only pivots flagged in rowAny fetch their full 1 KB mask row
//      (coalesced, lane k*32+lane layout) and OR it into rem[].
// Sparse overlaps => almost no memory on the sequential chain.
// =====================================================================
__global__ __launch_bounds__(32)
void nms_scan_kernel(const unsigned* __restrict__ keepInit,
                     const unsigned* __restrict__ mask,
                     const unsigned* __restrict__ diag,
                     const unsigned* __restrict__ rowAny,
                     unsigned* __restrict__ keepOut) {
    const int lane = threadIdx.x;           // 0..31
    unsigned rem[8];
#pragma unroll
    for (int k = 0; k < 8; ++k) rem[k] = 0u;

#pragma unroll
    for (int k = 0; k < 8; ++k) {           // word-group (register index)
        for (int s = 0; s < 32; ++s) {      // word within group
            const int w = k * 32 + s;       // current 32-box word (0..255)
            unsigned remW = (unsigned)__shfl((int)rem[k], s);
            unsigned di   = diag[(size_t)w * 32 + lane]; // lane's diag row
            unsigned raW  = rowAny[w];                   // uniform
            unsigned cur  = keepInit[w] & ~remW;         // survivors

            // ---- intra-word greedy resolution (registers + shuffles) ----
            unsigned rema = cur;
            unsigned keepBits = 0u;
            while (rema) {
                int b = __ffs(rema) - 1;                 // earliest survivor
                keepBits |= (1u << b);
                unsigned dw = (unsigned)__shfl((int)di, b); // bits > b only
                rema &= ~dw;
                rema &= ~(1u << b);
            }

            // ---- inter-word updates for pivots that reach other words ----
            unsigned need = keepBits & raW;
            while (need) {
                int b = __ffs(need) - 1;
                need &= need - 1;
                const size_t rowBase = (size_t)(w * 32 + b) * NWORDS;
#pragma unroll
                for (int k2 = 0; k2 < 8; ++k2)
                    rem[k2] |= mask[rowBase + (size_t)k2 * 32 + lane];
            }

            if (lane == 0) keepOut[w] = keepBits;
        }
    }
}

// =====================================================================
// K6: write (N,6) output:  [box * SCALER * kf, conf * kf]
// =====================================================================
__global__ void nms_output_kernel(const float* __restrict__ sBoxes,
                                  const float* __restrict__ sConf,
                                  const unsigned* __restrict__ keep,
                                  float* __restrict__ out) {
    int i = blockIdx.x * 256 + threadIdx.x;
    float kf = ((keep[i >> 5] >> (i & 31)) & 1u) ? 1.0f : 0.0f;
    float4 b = ((const float4*)sBoxes)[i];
    out[i * 6 + 0] = b.x * SCALER_F * kf;
    out[i * 6 + 1] = b.y * SCALER_F * kf;
    out[i * 6 + 2] = b.z * SCALER_F * kf;
    out[i * 6 + 3] = b.w * SCALER_F * kf;
    out[i * 6 + 4] = sConf[2 * i + 0] * kf;
    out[i * 6 + 5] = sConf[2 * i + 1] * kf;
}

// =====================================================================
// launcher
// =====================================================================
extern "C" void kernel_launch(void* const* d_in, const int* in_sizes, int n_in,
                              void* d_out, int out_size, void* d_ws, size_t ws_size,
                              hipStream_t stream) {
    (void)in_sizes; (void)n_in; (void)out_size; (void)ws_size;
    const float* cls_conf = (const float*)d_in[0];   // (N,2) f32
    const float* bboxes   = (const float*)d_in[1];   // (N,4) f32
    float* out = (float*)d_out;                      // (N,6) f32

    // workspace layout (16B-aligned offsets; total ~8.7 MB)
    char* ws = (char*)d_ws;
    float*    ms       = (float*)(ws + 0);           //  32 KB masked scores
    int*      order    = (int*)(ws + 32768);         //  32 KB sorted indices
    float*    sBoxes   = (float*)(ws + 65536);       // 128 KB sorted boxes
    float*    sConf    = (float*)(ws + 196608);      //  64 KB sorted conf
    unsigned* keepInit = (unsigned*)(ws + 262144);   //   1 KB
    unsigned* keepFin  = (unsigned*)(ws + 263168);   //   1 KB
    unsigned* rowAny   = (unsigned*)(ws + 264192);   //   1 KB
    unsigned* diag     = (unsigned*)(ws + 265216);   //  32 KB own-word rows
    unsigned* mask     = (unsigned*)(ws + 297984);   //   8 MB suppression bits

    nms_score_kernel <<<N_BOX / 256, 256, 0, stream>>>(cls_conf, ms);
    nms_sort_kernel  <<<1, 1024, 0, stream>>>(ms, order);
    nms_gather_kernel<<<N_BOX / 256, 256, 0, stream>>>(cls_conf, bboxes, order,
                                                       sBoxes, sConf, keepInit, rowAny);
    nms_mask_kernel  <<<dim3(NWORDS, N_BOX / 256), 256, 0, stream>>>(sBoxes, mask,
                                                                     diag, rowAny);
    nms_scan_kernel  <<<1, 32, 0, stream>>>(keepInit, mask, diag, rowAny, keepFin);
    nms_output_kernel<<<N_BOX / 256, 256, 0, stream>>>(sBoxes, sConf, keepFin, out);
}